// SteamGNN_81724637708444
// MI455X (gfx1250) — compile-verified
//
#include <hip/hip_runtime.h>

typedef __attribute__((ext_vector_type(2))) float v2f;
typedef __attribute__((ext_vector_type(8))) float v8f;

#define DH 128

// ---------------------------------------------------------------------------
// Zero a float buffer (agg / deg scratch must be re-zeroed every call since
// the harness does not re-poison between replays).
// ---------------------------------------------------------------------------
__global__ void zero_kernel(float* __restrict__ buf, int n) {
    int i = blockIdx.x * blockDim.x + threadIdx.x;
    if (i < n) buf[i] = 0.0f;
}

// ---------------------------------------------------------------------------
// Edge scatter: agg[dst] += h[src] (128 channels), deg[dst] += 1.
// 32 threads per edge, 4 channels each -> fully coalesced float4 row reads,
// f32 atomics resolved at L2 (whole agg buffer is L2-resident on MI455X).
// ---------------------------------------------------------------------------
__global__ void scatter_kernel(const float* __restrict__ h,
                               const int*   __restrict__ src,
                               const int*   __restrict__ dst,
                               float* __restrict__ agg,
                               float* __restrict__ deg,   // null for layer 2
                               int nE) {
    long long t = (long long)blockIdx.x * blockDim.x + threadIdx.x;
    int e  = (int)(t >> 5);
    int cg = (int)(t & 31);
    if (e >= nE) return;
    int s = src[e];
    int d = dst[e];
    const float4 val = *(const float4*)(h + (size_t)s * DH + cg * 4);
    float* o = agg + (size_t)d * DH + cg * 4;
    atomicAdd(o + 0, val.x);
    atomicAdd(o + 1, val.y);
    atomicAdd(o + 2, val.z);
    atomicAdd(o + 3, val.w);
    if (deg != nullptr && cg == 0) atomicAdd(deg + d, 1.0f);
}

// ---------------------------------------------------------------------------
// Fused SAGE layer: out = relu( (agg/max(deg,1)) @ Wl + b + xin @ Wr )
// One wave32 per 16x16 output tile; both GEMMs accumulate into one C fragment
// via V_WMMA_F32_16X16X4_F32 (full fp32 precision, matches reference).
// Block = 8 waves = one tileM row covering all 8 tileN column tiles.
// ---------------------------------------------------------------------------
__global__ __launch_bounds__(256) void sage_layer_kernel(
    const float* __restrict__ xin,
    const float* __restrict__ agg,
    const float* __restrict__ deg,
    const float* __restrict__ Wl,    // [128 x 128], row-major (k, n)
    const float* __restrict__ Wr,    // [128 x 128], row-major (k, n)
    const float* __restrict__ bias,  // [128]
    float* __restrict__ out,         // [nRows x 128]
    int nRows)
{
    const int wavesPerBlock = blockDim.x >> 5;
    const int waveId = blockIdx.x * wavesPerBlock + (threadIdx.x >> 5);
    const int lane   = threadIdx.x & 31;

    const int tilesN = DH / 16;            // 8
    const int tileM  = waveId / tilesN;
    const int tileN  = waveId % tilesN;
    if (tileM * 16 >= nRows) return;       // wave-uniform: EXEC stays all-1s

    // A-fragment addressing (16x4 f32): M = lane%16, K-pair = 2*(lane/16)
    const int rowA = tileM * 16 + (lane & 15);
    const int row  = (rowA < nRows) ? rowA : (nRows - 1);   // clamp (N%16==0 here)
    const int koff = (lane >> 4) << 1;     // 0 or 2
    // B-fragment addressing (4x16 f32): N = lane%16, same K split
    const int col  = tileN * 16 + (lane & 15);

    const float dv  = deg[row];
    const float inv = 1.0f / fmaxf(dv, 1.0f);

    const float* aggRow = agg + (size_t)row * DH;
    const float* xRow   = xin + (size_t)row * DH;

    v8f c = {};
    #pragma unroll
    for (int k0 = 0; k0 < DH; k0 += 4) {
        const int k = k0 + koff;
        v2f a_mean = *(const v2f*)(aggRow + k);   // 8B-aligned (k even)
        a_mean.x *= inv;
        a_mean.y *= inv;
        v2f a_self = *(const v2f*)(xRow + k);
        v2f b_l, b_r;
        b_l.x = Wl[(size_t)(k + 0) * DH + col];
        b_l.y = Wl[(size_t)(k + 1) * DH + col];
        b_r.x = Wr[(size_t)(k + 0) * DH + col];
        b_r.y = Wr[(size_t)(k + 1) * DH + col];
        // D = A*B + C, fp32 WMMA (16x16x4)
        c = __builtin_amdgcn_wmma_f32_16x16x4_f32(false, a_mean, false, b_l,
                                                  (short)0, c, false, false);
        c = __builtin_amdgcn_wmma_f32_16x16x4_f32(false, a_self, false, b_r,
                                                  (short)0, c, false, false);
    }

    // C/D layout: lanes 0-15 hold M = tile*16 + r, lanes 16-31 hold M = +8
    const float bcol  = bias[col];
    const int   mbase = tileM * 16 + ((lane >> 4) << 3);
    #pragma unroll
    for (int r = 0; r < 8; ++r) {
        int m = mbase + r;
        if (m < nRows) {
            float v = c[r] + bcol;
            out[(size_t)m * DH + col] = fmaxf(v, 0.0f);
        }
    }
}

// ---------------------------------------------------------------------------
// Edge decode: out[p] = dot(h[u[p]], h[v[p]]) over 128 channels.
// One wave32 per pair: float4 per lane, wave shuffle reduction.
// ---------------------------------------------------------------------------
__global__ void decode_kernel(const float* __restrict__ h,
                              const int*   __restrict__ u,
                              const int*   __restrict__ v,
                              float* __restrict__ out,
                              int nP)
{
    int wid  = blockIdx.x * (blockDim.x >> 5) + (threadIdx.x >> 5);
    int lane = threadIdx.x & 31;
    if (wid >= nP) return;
    int a = u[wid];
    int b = v[wid];
    const float4 ha = *(const float4*)(h + (size_t)a * DH + lane * 4);
    const float4 hb = *(const float4*)(h + (size_t)b * DH + lane * 4);
    float s = ha.x * hb.x + ha.y * hb.y + ha.z * hb.z + ha.w * hb.w;
    #pragma unroll
    for (int off = 16; off >= 1; off >>= 1)
        s += __shfl_xor(s, off, 32);
    if (lane == 0) out[wid] = s;
}

// ---------------------------------------------------------------------------
extern "C" void kernel_launch(void* const* d_in, const int* in_sizes, int n_in,
                              void* d_out, int out_size, void* d_ws, size_t ws_size,
                              hipStream_t stream) {
    const float* x   = (const float*)d_in[0];
    const int*   ei  = (const int*)  d_in[1];
    const int*   ep  = (const int*)  d_in[2];
    const float* W1l = (const float*)d_in[3];
    const float* b1  = (const float*)d_in[4];
    const float* W1r = (const float*)d_in[5];
    const float* W2l = (const float*)d_in[6];
    const float* b2  = (const float*)d_in[7];
    const float* W2r = (const float*)d_in[8];
    float* out = (float*)d_out;

    const int N = in_sizes[0] / DH;
    const int E = in_sizes[1] / 2;
    const int P = in_sizes[2] / 2;

    const int* src = ei;
    const int* dst = ei + E;
    const int* pu  = ep;
    const int* pv  = ep + P;

    // Workspace layout: [ agg (N*128) | deg (N) | h1 (N*128) | h2 (N*128) ]
    float* agg = (float*)d_ws;
    float* deg = agg + (size_t)N * DH;
    float* h1  = deg + N;
    float* h2  = h1 + (size_t)N * DH;

    const int tilesM = (N + 15) / 16;
    const int waves  = tilesM * (DH / 16);
    const long long scatterT = (long long)E * 32;

    // ---- Layer 1 ----
    {
        int n = N * DH + N;  // agg + deg contiguous
        zero_kernel<<<(n + 255) / 256, 256, 0, stream>>>(agg, n);
    }
    scatter_kernel<<<(int)((scatterT + 255) / 256), 256, 0, stream>>>(
        x, src, dst, agg, deg, E);
    sage_layer_kernel<<<(waves + 7) / 8, 256, 0, stream>>>(
        x, agg, deg, W1l, W1r, b1, h1, N);

    // ---- Layer 2 ----
    {
        int n = N * DH;
        zero_kernel<<<(n + 255) / 256, 256, 0, stream>>>(agg, n);
    }
    scatter_kernel<<<(int)((scatterT + 255) / 256), 256, 0, stream>>>(
        h1, src, dst, agg, nullptr, E);
    sage_layer_kernel<<<(waves + 7) / 8, 256, 0, stream>>>(
        h1, agg, deg, W2l, W2r, b2, h2, N);

    // ---- Decode ----
    decode_kernel<<<(P + 7) / 8, 256, 0, stream>>>(h2, pu, pv, out, P);
}